// ConditioningEncoder_35631048687803
// MI455X (gfx1250) — compile-verified
//
#include <hip/hip_runtime.h>

typedef __attribute__((ext_vector_type(2))) float v2f;
typedef __attribute__((ext_vector_type(8))) float v8f;

#define ROWS 64          // rows (positions) per workgroup = 4 WMMA M-tiles
#define MT   4           // M-tiles per wave
#define KDIM 320
#define NDIM 256
#define BLOCK 256        // 8 waves

__global__ __launch_bounds__(BLOCK)
void cond_encoder_kernel(const float* __restrict__ f0,
                         const int*   __restrict__ phone,
                         const float* __restrict__ dur,
                         const int*   __restrict__ midi,
                         const float* __restrict__ f0_w1, const float* __restrict__ f0_b1,
                         const float* __restrict__ f0_w2, const float* __restrict__ f0_b2,
                         const float* __restrict__ phone_tab, const float* __restrict__ midi_tab,
                         const float* __restrict__ dur_w1, const float* __restrict__ dur_b1,
                         const float* __restrict__ dur_w2, const float* __restrict__ dur_b2,
                         const float* __restrict__ w1, const float* __restrict__ b1,
                         const float* __restrict__ ln_g, const float* __restrict__ ln_b,
                         const float* __restrict__ w2, const float* __restrict__ b2,
                         float* __restrict__ out)
{
    __shared__ float lds_comb[ROWS][KDIM];   // 64x320 concat features (80 KB)
    __shared__ float lds_h[ROWS][NDIM];      // 64x256 hidden (64 KB)
    __shared__ float lds_red[ROWS][8];       // LN partials: [0:4) sums, [4:8) sumsq
    __shared__ float lds_mu[ROWS];
    __shared__ float lds_rs[ROWS];

    // f0/dur hidden scratch aliased into lds_h (consumed before lds_h is written,
    // with a __syncthreads() in between)
    float (*lds_hid)[64] = (float (*)[64])(&lds_h[0][0]);   // 64x64 region (16 KB)

    const int tid = threadIdx.x;
    const long long rowbase = (long long)blockIdx.x * ROWS;

    // ---- Phase A: f0/dur hidden layers (scalar -> 32, ReLU) ----
    for (int i = tid; i < ROWS * 32; i += BLOCK) {
        const int r = i >> 5, j = i & 31;
        float x = f0[rowbase + r];
        float h = x * f0_w1[j] + f0_b1[j];
        lds_hid[r][j] = h > 0.f ? h : 0.f;
        float y = dur[rowbase + r];
        float g = y * dur_w1[j] + dur_b1[j];
        lds_hid[r][32 + j] = g > 0.f ? g : 0.f;
    }
    __syncthreads();

    // ---- Phase B: build 64x320 combined tile in LDS ----
    // per row: [0:64) f0_emb | [64:192) phone_emb | [192:256) midi_emb | [256:320) dur_emb
    for (int i = tid; i < ROWS * 64; i += BLOCK) {
        const int r = i >> 6, k = i & 63;
        float acc = f0_b2[k];
        #pragma unroll 8
        for (int j = 0; j < 32; ++j) acc += lds_hid[r][j] * f0_w2[j * 64 + k];
        lds_comb[r][k] = acc;
        float acc2 = dur_b2[k];
        #pragma unroll 8
        for (int j = 0; j < 32; ++j) acc2 += lds_hid[r][32 + j] * dur_w2[j * 64 + k];
        lds_comb[r][256 + k] = acc2;
        const int mi = midi[rowbase + r];
        lds_comb[r][192 + k] = midi_tab[mi * 64 + k];
    }
    for (int i = tid; i < ROWS * 128; i += BLOCK) {
        const int r = i >> 7, k = i & 127;
        const int ph = phone[rowbase + r];
        lds_comb[r][64 + k] = phone_tab[ph * 128 + k];
    }
    __syncthreads();

    // ---- GEMM1: [64x320] @ [320x256] via v_wmma_f32_16x16x4_f32 ----
    // Each wave: 2 N-tiles (32 cols) x 4 M-tiles; B fragments reused across M-tiles.
    const int wave = tid >> 5;
    const int lane = tid & 31;
    const int m    = lane & 15;            // row within M-tile / col within C-tile
    const int koff = (lane >> 4) << 1;     // lanes 0-15 -> K+0/K+1, lanes 16-31 -> K+2/K+3
    const int nc0  = wave * 32 + m;
    const int nc1  = nc0 + 16;

    v8f acc[MT][2];
    #pragma unroll
    for (int mt = 0; mt < MT; ++mt) { acc[mt][0] = (v8f){}; acc[mt][1] = (v8f){}; }

    for (int kk = 0; kk < KDIM; kk += 4) {
        v2f bv0, bv1;
        bv0.x = w1[(kk + koff)     * NDIM + nc0];
        bv0.y = w1[(kk + koff + 1) * NDIM + nc0];
        bv1.x = w1[(kk + koff)     * NDIM + nc1];
        bv1.y = w1[(kk + koff + 1) * NDIM + nc1];
        #pragma unroll
        for (int mt = 0; mt < MT; ++mt) {
            v2f a;
            a.x = lds_comb[mt * 16 + m][kk + koff];
            a.y = lds_comb[mt * 16 + m][kk + koff + 1];
            acc[mt][0] = __builtin_amdgcn_wmma_f32_16x16x4_f32(false, a, false, bv0, (short)0, acc[mt][0], false, false);
            acc[mt][1] = __builtin_amdgcn_wmma_f32_16x16x4_f32(false, a, false, bv1, (short)0, acc[mt][1], false, false);
        }
    }
    // C/D layout: VGPR v -> M = v (+8 for lanes 16-31), N = lane&15
    {
        const int rbase = (lane >> 4) << 3;
        const float bias0 = b1[nc0];
        const float bias1 = b1[nc1];
        #pragma unroll
        for (int mt = 0; mt < MT; ++mt) {
            #pragma unroll
            for (int v = 0; v < 8; ++v) {
                const int r = mt * 16 + rbase + v;
                lds_h[r][nc0] = acc[mt][0][v] + bias0;
                lds_h[r][nc1] = acc[mt][1][v] + bias1;
            }
        }
    }
    __syncthreads();

    // ---- LayerNorm(256) + ReLU ----
    {
        const int r = tid >> 2, s = tid & 3;       // 4 segments of 64 per row
        float sum = 0.f, sq = 0.f;
        #pragma unroll 16
        for (int j = 0; j < 64; ++j) {
            float x = lds_h[r][s * 64 + j];
            sum += x; sq += x * x;
        }
        lds_red[r][s] = sum;
        lds_red[r][4 + s] = sq;
    }
    __syncthreads();
    if (tid < ROWS) {
        float sum = 0.f, sq = 0.f;
        #pragma unroll
        for (int s = 0; s < 4; ++s) { sum += lds_red[tid][s]; sq += lds_red[tid][4 + s]; }
        const float mu  = sum * (1.f / 256.f);
        const float var = sq * (1.f / 256.f) - mu * mu;
        lds_mu[tid] = mu;
        lds_rs[tid] = rsqrtf(var + 1e-5f);
    }
    __syncthreads();
    for (int i = tid; i < ROWS * NDIM; i += BLOCK) {
        const int r = i >> 8, c = i & 255;
        float x = (lds_h[r][c] - lds_mu[r]) * lds_rs[r] * ln_g[c] + ln_b[c];
        lds_h[r][c] = x > 0.f ? x : 0.f;
    }
    __syncthreads();

    // ---- GEMM2: [64x256] @ [256x256] ----
    v8f d[MT][2];
    #pragma unroll
    for (int mt = 0; mt < MT; ++mt) { d[mt][0] = (v8f){}; d[mt][1] = (v8f){}; }

    for (int kk = 0; kk < NDIM; kk += 4) {
        v2f bv0, bv1;
        bv0.x = w2[(kk + koff)     * NDIM + nc0];
        bv0.y = w2[(kk + koff + 1) * NDIM + nc0];
        bv1.x = w2[(kk + koff)     * NDIM + nc1];
        bv1.y = w2[(kk + koff + 1) * NDIM + nc1];
        #pragma unroll
        for (int mt = 0; mt < MT; ++mt) {
            v2f a;
            a.x = lds_h[mt * 16 + m][kk + koff];
            a.y = lds_h[mt * 16 + m][kk + koff + 1];
            d[mt][0] = __builtin_amdgcn_wmma_f32_16x16x4_f32(false, a, false, bv0, (short)0, d[mt][0], false, false);
            d[mt][1] = __builtin_amdgcn_wmma_f32_16x16x4_f32(false, a, false, bv1, (short)0, d[mt][1], false, false);
        }
    }
    {
        const int rbase = (lane >> 4) << 3;
        const float bias0 = b2[nc0];
        const float bias1 = b2[nc1];
        #pragma unroll
        for (int mt = 0; mt < MT; ++mt) {
            #pragma unroll
            for (int v = 0; v < 8; ++v) {
                const int r = mt * 16 + rbase + v;
                const long long o = (rowbase + r) * NDIM;
                out[o + nc0] = d[mt][0][v] + bias0;
                out[o + nc1] = d[mt][1][v] + bias1;
            }
        }
    }
}

extern "C" void kernel_launch(void* const* d_in, const int* in_sizes, int n_in,
                              void* d_out, int out_size, void* d_ws, size_t ws_size,
                              hipStream_t stream) {
    const float* f0        = (const float*)d_in[0];
    const int*   phone     = (const int*)  d_in[1];
    const float* dur       = (const float*)d_in[2];
    const int*   midi      = (const int*)  d_in[3];
    const float* f0_w1     = (const float*)d_in[4];
    const float* f0_b1     = (const float*)d_in[5];
    const float* f0_w2     = (const float*)d_in[6];
    const float* f0_b2     = (const float*)d_in[7];
    const float* phone_tab = (const float*)d_in[8];
    const float* midi_tab  = (const float*)d_in[9];
    const float* dur_w1    = (const float*)d_in[10];
    const float* dur_b1    = (const float*)d_in[11];
    const float* dur_w2    = (const float*)d_in[12];
    const float* dur_b2    = (const float*)d_in[13];
    const float* w1        = (const float*)d_in[14];
    const float* b1        = (const float*)d_in[15];
    const float* ln_g      = (const float*)d_in[16];
    const float* ln_b      = (const float*)d_in[17];
    const float* w2        = (const float*)d_in[18];
    const float* b2        = (const float*)d_in[19];
    float* out = (float*)d_out;

    const int npos   = in_sizes[0];          // B*T = 65536
    const int blocks = npos / ROWS;          // 1024 blocks of 64 rows

    cond_encoder_kernel<<<blocks, BLOCK, 0, stream>>>(
        f0, phone, dur, midi,
        f0_w1, f0_b1, f0_w2, f0_b2,
        phone_tab, midi_tab,
        dur_w1, dur_b1, dur_w2, dur_b2,
        w1, b1, ln_g, ln_b, w2, b2, out);
}